// SSFE_45921790329318
// MI455X (gfx1250) — compile-verified
//
#include <hip/hip_runtime.h>
#include <hip/hip_bf16.h>

// ---------------------------------------------------------------------------
// SSFE pipeline for MI455X (gfx1250), fp32 end-to-end via V_WMMA_F32_16X16X4_F32
// B=8, C=128, H=W=64, P=4096, U=3
// ---------------------------------------------------------------------------

typedef float v2f __attribute__((ext_vector_type(2)));
typedef float v8f __attribute__((ext_vector_type(8)));

#define BB   8
#define CC   128
#define HH   64
#define WW   64
#define PP   (HH * WW)          // 4096
#define NBC  (BB * CC * PP)     // 4,194,304 floats
#define NBP  (BB * PP)          // 32,768 floats

#define EPS_NORM 1e-12f
#define EPS_COS  1e-8f
#define EPS_BN   1e-5f

// ---------------------------------------------------------------------------
// Generic channel GEMM: out[b,o,p] = sum_c Wm[o,c] * in[b,c,p]  (+epilogue)
// EPI: 0 = none, 1 = +bias, relu, 2 = +bias
// Block: 256 threads = 8 waves; wave w owns out-channels [16w,16w+16).
// Grid:  B * (P/16) = 2048 blocks; each block does one 16-position tile.
// Per-wave: 16x16 fp32 tile accumulated with 32x V_WMMA_F32_16X16X4_F32.
// ---------------------------------------------------------------------------
template <int EPI>
__global__ __launch_bounds__(256) void wmma_gemm128(
    const float* __restrict__ Wm, const float* __restrict__ in,
    const float* __restrict__ bias, float* __restrict__ out) {
  const int tile = blockIdx.x;
  const int b    = tile >> 8;        // P/16 = 256 tiles per batch
  const int pt   = tile & 255;
  const int p0   = pt * 16;

  const int lane = threadIdx.x & 31;
  const int wave = threadIdx.x >> 5;
  const int half = lane >> 4;        // 0: lanes 0-15, 1: lanes 16-31
  const int lidx = lane & 15;
  const int obase = wave * 16;

  const float* inb = in + (size_t)b * CC * PP;
  const int pcol = p0 + lidx;                      // this lane's N (position)
  const int arow = (obase + lidx) * CC + 2 * half; // this lane's A row offset

  v8f acc = {};
#pragma unroll 8
  for (int k = 0; k < CC; k += 4) {
    // A: W tile (16 x 4). lanes0-15 -> K={k,k+1}; lanes16-31 -> K={k+2,k+3}
    v2f a = *(const v2f*)(Wm + arow + k);
    // B: input tile (4 x 16), same K split, N striped across lanes
    const int kk = k + 2 * half;
    v2f bv;
    bv.x = inb[(size_t)(kk + 0) * PP + pcol];
    bv.y = inb[(size_t)(kk + 1) * PP + pcol];
    acc = __builtin_amdgcn_wmma_f32_16x16x4_f32(false, a, false, bv,
                                                (short)0, acc, false, false);
  }

  // D layout: VGPR r -> (M = r, N = lane) for lanes 0-15, (M = r+8) for 16-31
#pragma unroll
  for (int r = 0; r < 8; ++r) {
    const int o = obase + r + 8 * half;
    float v = acc[r];
    if (EPI >= 1) v += bias[o];
    if (EPI == 1) v = fmaxf(v, 0.0f);
    out[((size_t)b * CC + o) * PP + pcol] = v;
  }
}

// ---------------------------------------------------------------------------
// Self-similarity conv GEMM:
//   y[b,o,p] = bc[o] + sum_{du,dv,i} Wc[o,i,du,dv] * relu(q[b,i,p]*qpad[b,i,p_shift])
// 9 accumulated K=128 GEMM passes, B-tiles computed on the fly.
// Halo handling is branch-free: always load from a clamped (in-range) address
// and multiply by a 0/1 mask -> v_cndmask + plain load, no EXEC toggling
// around the WMMA stream.
// ---------------------------------------------------------------------------
__global__ __launch_bounds__(256) void wmma_selfsim_conv(
    const float* __restrict__ q, const float* __restrict__ Wc,
    const float* __restrict__ bc, float* __restrict__ y) {
  const int tile = blockIdx.x;
  const int b    = tile >> 8;
  const int pt   = tile & 255;
  const int p0   = pt * 16;

  const int lane = threadIdx.x & 31;
  const int wave = threadIdx.x >> 5;
  const int half = lane >> 4;
  const int lidx = lane & 15;
  const int obase = wave * 16;

  const int hrow = p0 >> 6;              // image row of this tile (uniform)
  const int wcol = (p0 & 63) + lidx;     // image col for this lane
  const float* qb = q + (size_t)b * CC * PP;
  const int pcol = p0 + lidx;

  // Precompute clamped shifted positions + masks for the 9 taps (loop-invariant).
  int   psafe[9];
  float msk[9];
#pragma unroll
  for (int du = 0; du < 3; ++du) {
    const int hs = hrow + du - 1;
    const bool hok = ((unsigned)hs < (unsigned)HH);
#pragma unroll
    for (int dv = 0; dv < 3; ++dv) {
      const int ws = wcol + dv - 1;
      const bool ok = hok && ((unsigned)ws < (unsigned)WW);
      psafe[du * 3 + dv] = ok ? (hs * WW + ws) : pcol;  // clamped, always in-range
      msk[du * 3 + dv]   = ok ? 1.0f : 0.0f;
    }
  }

  v8f acc = {};
#pragma unroll 2
  for (int k = 0; k < CC; k += 4) {
    const int kk = k + 2 * half;
    const float* qk0 = qb + (size_t)(kk + 0) * PP;
    const float* qk1 = qb + (size_t)(kk + 1) * PP;
    const float qc0 = qk0[pcol];
    const float qc1 = qk1[pcol];
    const float* wrow = Wc + ((obase + lidx) * CC + kk) * 9;
#pragma unroll
    for (int t = 0; t < 9; ++t) {
      const float qs0 = qk0[psafe[t]] * msk[t];
      const float qs1 = qk1[psafe[t]] * msk[t];
      v2f bv;
      bv.x = fmaxf(qc0 * qs0, 0.0f);
      bv.y = fmaxf(qc1 * qs1, 0.0f);
      v2f a;
      a.x = wrow[t];        // (o, i=kk,   tap t)
      a.y = wrow[t + 9];    // (o, i=kk+1, tap t)
      acc = __builtin_amdgcn_wmma_f32_16x16x4_f32(false, a, false, bv,
                                                  (short)0, acc, false, false);
    }
  }
#pragma unroll
  for (int r = 0; r < 8; ++r) {
    const int o = obase + r + 8 * half;
    y[((size_t)b * CC + o) * PP + pcol] = acc[r] + bc[o];
  }
}

// ---------------------------------------------------------------------------
// Channel-wise L2 normalize (in place); pnorm = post-normalization norm.
// One thread per (b,p); consecutive threads -> consecutive p (coalesced).
// ---------------------------------------------------------------------------
__global__ __launch_bounds__(256) void normalize_q(float* __restrict__ q,
                                                   float* __restrict__ pnorm) {
  const int idx = blockIdx.x * 256 + threadIdx.x;   // 0..B*P
  const int b = idx >> 12;
  const int p = idx & (PP - 1);
  float* qp = q + (size_t)b * CC * PP + p;
  float s = 0.0f;
  for (int c = 0; c < CC; ++c) {
    const float v = qp[(size_t)c * PP];
    s += v * v;
  }
  const float n = sqrtf(s);
  const float inv = 1.0f / fmaxf(n, EPS_NORM);
  for (int c = 0; c < CC; ++c) qp[(size_t)c * PP] *= inv;
  pnorm[idx] = n * inv;   // == n / max(n, EPS_NORM)
}

// ---------------------------------------------------------------------------
// Cosine similarity with the center pixel: sim[b,p] = clip(num/den, 0).
// ---------------------------------------------------------------------------
__global__ __launch_bounds__(256) void center_sim(const float* __restrict__ q,
                                                  const float* __restrict__ pnorm,
                                                  float* __restrict__ sim) {
  const int idx = blockIdx.x * 256 + threadIdx.x;
  const int b = idx >> 12;
  const int p = idx & (PP - 1);
  const int pc = (PP - 1) / 2;   // 2047
  const float* qb = q + (size_t)b * CC * PP;
  float dot = 0.0f;
  for (int c = 0; c < CC; ++c)
    dot += qb[(size_t)c * PP + p] * qb[(size_t)c * PP + pc];
  const float den = fmaxf(pnorm[b * PP + pc], EPS_COS) *
                    fmaxf(pnorm[b * PP + p], EPS_COS);
  sim[idx] = fmaxf(dot / den, 0.0f);
}

// ---------------------------------------------------------------------------
// BN statistics: one block per channel; mean and var over (b, p).
// ---------------------------------------------------------------------------
__global__ __launch_bounds__(256) void bn_stats(const float* __restrict__ y,
                                                float* __restrict__ stats) {
  const int c = blockIdx.x;
  float s = 0.0f, s2 = 0.0f;
  for (int i = threadIdx.x; i < BB * PP; i += 256) {
    const int b = i >> 12;
    const int p = i & (PP - 1);
    const float v = y[((size_t)b * CC + c) * PP + p];
    s += v;
    s2 += v * v;
  }
  __shared__ float sh[256], sh2[256];
  sh[threadIdx.x] = s;
  sh2[threadIdx.x] = s2;
  __syncthreads();
  for (int o = 128; o > 0; o >>= 1) {
    if (threadIdx.x < o) {
      sh[threadIdx.x]  += sh[threadIdx.x + o];
      sh2[threadIdx.x] += sh2[threadIdx.x + o];
    }
    __syncthreads();
  }
  if (threadIdx.x == 0) {
    const float inv = 1.0f / (float)(BB * PP);
    const float mean = sh[0] * inv;
    stats[c]      = mean;
    stats[CC + c] = sh2[0] * inv - mean * mean;   // jnp.var (biased)
  }
}

// ---------------------------------------------------------------------------
// Fuse: f1 = x + relu(BN(y)) + sim*q   (written in place over y's region)
// ---------------------------------------------------------------------------
__global__ __launch_bounds__(256) void fuse_residual(
    const float* __restrict__ x, float* __restrict__ y /* -> f1 in place */,
    const float* __restrict__ q, const float* __restrict__ sim,
    const float* __restrict__ stats, const float* __restrict__ gamma,
    const float* __restrict__ beta) {
  const int idx = blockIdx.x * 256 + threadIdx.x;   // 0..B*C*P
  const int p = idx & (PP - 1);
  const int c = (idx >> 12) & (CC - 1);
  const int b = idx >> 19;
  const float mean = stats[c];
  const float var  = stats[CC + c];
  const float bn = gamma[c] * (y[idx] - mean) * rsqrtf(var + EPS_BN) + beta[c];
  y[idx] = x[idx] + fmaxf(bn, 0.0f) + sim[b * PP + p] * q[idx];
}

// ---------------------------------------------------------------------------
// Host-side launch
// ---------------------------------------------------------------------------
extern "C" void kernel_launch(void* const* d_in, const int* in_sizes, int n_in,
                              void* d_out, int out_size, void* d_ws, size_t ws_size,
                              hipStream_t stream) {
  const float* x     = (const float*)d_in[0];
  const float* Wq    = (const float*)d_in[1];
  const float* Wc    = (const float*)d_in[2];
  const float* bc    = (const float*)d_in[3];
  const float* gamma = (const float*)d_in[4];
  const float* beta  = (const float*)d_in[5];
  const float* W1    = (const float*)d_in[6];
  const float* b1    = (const float*)d_in[7];
  const float* W2    = (const float*)d_in[8];
  const float* b2    = (const float*)d_in[9];
  float* out = (float*)d_out;

  float* ws    = (float*)d_ws;
  float* q     = ws;               // NBC floats; reused as t after fuse
  float* y     = ws + NBC;         // NBC floats; becomes f1 in place
  float* pnorm = ws + 2 * NBC;     // NBP
  float* sim   = pnorm + NBP;      // NBP
  float* stats = sim + NBP;        // 2*CC

  const int gemm_blocks = BB * (PP / 16);   // 2048
  const dim3 blk(256);

  // 1) q = Wq @ x
  wmma_gemm128<0><<<gemm_blocks, blk, 0, stream>>>(Wq, x, nullptr, q);
  // 2) normalize q along channels (in place), record per-position norm
  normalize_q<<<NBP / 256, blk, 0, stream>>>(q, pnorm);
  // 3) cosine similarity against center pixel
  center_sim<<<NBP / 256, blk, 0, stream>>>(q, pnorm, sim);
  // 4) y = Wc (*) relu(q * shift(q)) + bc   (9 accumulated WMMA GEMM passes)
  wmma_selfsim_conv<<<gemm_blocks, blk, 0, stream>>>(q, Wc, bc, y);
  // 5) per-channel BN statistics
  bn_stats<<<CC, blk, 0, stream>>>(y, stats);
  // 6) f1 = x + relu(BN(y)) + sim*q   (in place over y)
  fuse_residual<<<NBC / 256, blk, 0, stream>>>(x, y, q, sim, stats, gamma, beta);
  // 7) t = relu(W1 @ f1 + b1)   (into q's region)
  wmma_gemm128<1><<<gemm_blocks, blk, 0, stream>>>(W1, y, b1, q);
  // 8) out = W2 @ t + b2
  wmma_gemm128<2><<<gemm_blocks, blk, 0, stream>>>(W2, q, b2, out);
}